// Drift_32495722561662
// MI455X (gfx1250) — compile-verified
//
#include <hip/hip_runtime.h>
#include <math.h>

// MI455X / gfx1250, wave32. fp32 WMMA 16x16x4 path + hardware v_tanh_f32.
typedef __attribute__((ext_vector_type(2))) float v2f;
typedef __attribute__((ext_vector_type(8))) float v8f;

#define DEPTH 4
#define H1 64
#define H2 32
#define NP 64
#define DIMX 3
#define FSIZE 160   // 2*H1 + H2
#define BATCHN 256

// D = A(16x4) * B(4x16) + C, fp32 accumulate.
// 8-arg form: (neg_a, A, neg_b, B, c_mod, C, reuse_a, reuse_b)
__device__ __forceinline__ v8f wmma_f32_k4(v2f a, v2f b, v8f c) {
  return __builtin_amdgcn_wmma_f32_16x16x4_f32(false, a, false, b, (short)0, c,
                                               false, false);
}

// Hardware tanh (gfx1250 TRANS32 op). ~25x fewer VALU issues than libm tanhf,
// which dominates this workload (~100M tanh evals vs ~2.8M wmma issues).
__device__ __forceinline__ float fast_tanh(float x) {
#if __has_builtin(__builtin_amdgcn_tanhf)
  return __builtin_amdgcn_tanhf(x);
#else
  float r;
  // TRANS op: 1 independent op / V_NOP required before result use (ISA 7.4);
  // the compiler can't see through asm, so insert the v_nop ourselves.
  asm volatile("v_tanh_f32 %0, %1\n\tv_nop" : "=v"(r) : "v"(x));
  return r;
#endif
}

// ---------------------------------------------------------------------------
// Kernel 1: one block per (b, i). Runs the h2 chain for all 64 j-rows in LDS:
//   feat5 = [rij, |rij|, t]  ->  h = tanh(W0 feat5 + b0)          (VALU, K=5)
//   h = tanh(W1 h + b1) + h ; h = tanh(W2 h + b2) + h             (WMMA 32x32)
// Emits mean_j of [rij,|rij|] and of h after each stage (mfeat, m0, m1, m2).
// Never touches HBM with the 128MB h2 tensor.
// ---------------------------------------------------------------------------
__global__ __launch_bounds__(64) void drift_pair_chain(
    const float* __restrict__ x, const float* __restrict__ t,
    const float* __restrict__ fc2_w0, const float* __restrict__ fc2_b0,
    const float* __restrict__ fc2_w, const float* __restrict__ fc2_b,
    float* __restrict__ mfeat, float* __restrict__ m0,
    float* __restrict__ m1, float* __restrict__ m2) {
  const int i = blockIdx.x;
  const int b = blockIdx.y;
  const int tid = threadIdx.x;  // 0..63, two wave32s

  __shared__ float hA[NP][H2 + 4];   // pad->36: conflict-free strided reads
  __shared__ float hB[NP][H2 + 4];
  __shared__ float wS[H2][H2 + 4];
  __shared__ float sfeat[NP][4];
  __shared__ float biasS[H2];

  __builtin_prefetch(fc2_w, 0, 1);

  const float tval = t[0];
  const float xi0 = x[(b * NP + i) * 3 + 0];
  const float xi1 = x[(b * NP + i) * 3 + 1];
  const float xi2 = x[(b * NP + i) * 3 + 2];

  // --- per-thread: features + first fc2 layer (K=5, VALU) ---
  const int j = tid;
  float r0 = xi0 - x[(b * NP + j) * 3 + 0];
  float r1 = xi1 - x[(b * NP + j) * 3 + 1];
  float r2 = xi2 - x[(b * NP + j) * 3 + 2];
  if (j == i) { r0 += 1.f; r1 += 1.f; r2 += 1.f; }  // eye offset on diagonal
  const float nr = sqrtf(r0 * r0 + r1 * r1 + r2 * r2);
  sfeat[j][0] = r0; sfeat[j][1] = r1; sfeat[j][2] = r2; sfeat[j][3] = nr;
  {
    const float f5[5] = {r0, r1, r2, nr, tval};
    #pragma unroll
    for (int o = 0; o < H2; ++o) {
      float acc = fc2_b0[o];
      #pragma unroll
      for (int k = 0; k < 5; ++k) acc += f5[k] * fc2_w0[o * 5 + k];
      hA[j][o] = fast_tanh(acc);
    }
  }
  __syncthreads();

  // means of raw pair features and of stage-0 h2
  if (tid < 4) {
    float s = 0.f;
    for (int jj = 0; jj < NP; ++jj) s += sfeat[jj][tid];
    mfeat[(b * NP + i) * 4 + tid] = s * (1.f / NP);
  }
  if (tid < H2) {
    float s = 0.f;
    for (int jj = 0; jj < NP; ++jj) s += hA[jj][tid];
    m0[(b * NP + i) * H2 + tid] = s * (1.f / NP);
  }

  // --- residual 32x32 layers via fp32 WMMA ---
  const int wave = tid >> 5;
  const int lane = tid & 31;
  const int half = lane >> 4;   // K-half for A/B operands, M-half for D
  const int lr = lane & 15;

  for (int L = 0; L < 2; ++L) {
    const float* Wl = fc2_w + L * H2 * H2;
    for (int idx = tid; idx < H2 * H2; idx += 64)
      wS[idx >> 5][idx & 31] = Wl[idx];
    if (tid < H2) biasS[tid] = fc2_b[L * H2 + tid];
    __syncthreads();

    float(*src)[H2 + 4] = (L == 0) ? hA : hB;
    float(*dst)[H2 + 4] = (L == 0) ? hB : hA;
    const int rowbase = wave * 32;  // each wave owns 32 j-rows -> 2 row tiles

    #pragma unroll
    for (int rt = 0; rt < 2; ++rt) {
      #pragma unroll
      for (int ct = 0; ct < 2; ++ct) {
        v8f acc = {};
        #pragma unroll
        for (int kk = 0; kk < 8; ++kk) {
          const int k0 = kk * 4 + half * 2;
          const int ar = rowbase + rt * 16 + lr;
          v2f a;  a.x = src[ar][k0];              a.y = src[ar][k0 + 1];
          v2f bv; bv.x = wS[ct * 16 + lr][k0];    bv.y = wS[ct * 16 + lr][k0 + 1];
          acc = wmma_f32_k4(a, bv, acc);  // D[m][n] += sum_k h[m][k]*W[n][k]
        }
        #pragma unroll
        for (int r = 0; r < 8; ++r) {
          const int row = rowbase + rt * 16 + r + half * 8;
          const int col = ct * 16 + lr;
          dst[row][col] = fast_tanh(acc[r] + biasS[col]) + src[row][col];
        }
      }
    }
    __syncthreads();

    float* mo = (L == 0) ? m1 : m2;
    if (tid < H2) {
      float s = 0.f;
      for (int jj = 0; jj < NP; ++jj) s += dst[jj][tid];
      mo[(b * NP + i) * H2 + tid] = s * (1.f / NP);
    }
    __syncthreads();
  }
}

// ---------------------------------------------------------------------------
// Kernel 2: one block per batch b. Runs the h1 chain for all 64 particles.
// combine(h1,h2) = [h1 | mean_n h1 | mean_j h2] is never materialized: the
// WMMA A-operand gathers each K-region (h1 LDS / broadcast mean / mprev LDS).
// ---------------------------------------------------------------------------
__global__ __launch_bounds__(64) void drift_particle_chain(
    const float* __restrict__ x, const float* __restrict__ t,
    const float* __restrict__ fc1_w0, const float* __restrict__ fc1_b0,
    const float* __restrict__ fc1_w, const float* __restrict__ fc1_b,
    const float* __restrict__ final_w, const float* __restrict__ final_b,
    const float* __restrict__ mfeat, const float* __restrict__ m0,
    const float* __restrict__ m1, const float* __restrict__ m2,
    float* __restrict__ out) {
  const int b = blockIdx.x;
  const int tid = threadIdx.x;  // 0..63

  __shared__ float hA[NP][H1 + 4];
  __shared__ float hB[NP][H1 + 4];
  __shared__ float mprev[NP][H2 + 4];
  __shared__ float meanh[H1];
  __shared__ float sfeat[NP][4];
  __shared__ float smf[4];
  __shared__ float biasS[H1];

  __builtin_prefetch(fc1_w, 0, 1);

  const float tval = t[0];
  const int i = tid;
  const float x0 = x[(b * NP + i) * 3 + 0];
  const float x1 = x[(b * NP + i) * 3 + 1];
  const float x2 = x[(b * NP + i) * 3 + 2];
  const float nr = sqrtf(x0 * x0 + x1 * x1 + x2 * x2);
  sfeat[i][0] = x0; sfeat[i][1] = x1; sfeat[i][2] = x2; sfeat[i][3] = nr;
  __syncthreads();
  if (tid < 4) {
    float s = 0.f;
    for (int jj = 0; jj < NP; ++jj) s += sfeat[jj][tid];
    smf[tid] = s * (1.f / NP);
  }
  __syncthreads();

  // --- d=0: ft(13) = [x,|x| | mean | mean_j h2feat | t], K=13 -> VALU ---
  {
    float ft[13];
    ft[0] = x0; ft[1] = x1; ft[2] = x2; ft[3] = nr;
    ft[4] = smf[0]; ft[5] = smf[1]; ft[6] = smf[2]; ft[7] = smf[3];
    ft[8]  = mfeat[(b * NP + i) * 4 + 0];
    ft[9]  = mfeat[(b * NP + i) * 4 + 1];
    ft[10] = mfeat[(b * NP + i) * 4 + 2];
    ft[11] = mfeat[(b * NP + i) * 4 + 3];
    ft[12] = tval;
    for (int o = 0; o < H1; ++o) {
      float acc = fc1_b0[o];
      #pragma unroll
      for (int k = 0; k < 13; ++k) acc += ft[k] * fc1_w0[o * 13 + k];
      hA[i][o] = fast_tanh(acc);
    }
  }
  __syncthreads();

  const int wave = tid >> 5;
  const int lane = tid & 31;
  const int half = lane >> 4;
  const int lr = lane & 15;

  // --- d = 1..3: h1 = tanh(combine @ W^T + b) + h1  (64x160 @ 160x64) ---
  for (int d = 1; d <= 3; ++d) {
    const float* Wl = fc1_w + (d - 1) * H1 * FSIZE;
    const float* ms = (d == 1) ? m0 : (d == 2) ? m1 : m2;
    float(*src)[H1 + 4] = (d & 1) ? hA : hB;
    float(*dst)[H1 + 4] = (d & 1) ? hB : hA;

    for (int idx = tid; idx < NP * H2; idx += 64)
      mprev[idx >> 5][idx & 31] = ms[b * NP * H2 + idx];
    biasS[tid] = fc1_b[(d - 1) * H1 + tid];
    {  // mean over particles of current h1 (64 cols, 64 threads)
      float s = 0.f;
      for (int jj = 0; jj < NP; ++jj) s += src[jj][tid];
      meanh[tid] = s * (1.f / NP);
    }
    __syncthreads();

    #pragma unroll
    for (int rt = 0; rt < 4; ++rt) {
      #pragma unroll
      for (int ct2 = 0; ct2 < 2; ++ct2) {
        const int ct = wave * 2 + ct2;  // wave-uniform tile split
        v8f acc = {};
        #pragma unroll 8
        for (int kk = 0; kk < 40; ++kk) {
          const int k0 = kk * 4 + half * 2;
          const int ar = rt * 16 + lr;
          v2f a;
          if (kk < 16) {        // K 0..63   : h1
            a.x = src[ar][k0];          a.y = src[ar][k0 + 1];
          } else if (kk < 32) { // K 64..127 : broadcast mean_n h1
            a.x = meanh[k0 - 64];       a.y = meanh[k0 - 63];
          } else {              // K 128..159: mean_j h2 (per particle)
            a.x = mprev[ar][k0 - 128];  a.y = mprev[ar][k0 - 127];
          }
          const float* wr = Wl + (ct * 16 + lr) * FSIZE + k0;  // L2-resident
          v2f bv; bv.x = wr[0]; bv.y = wr[1];
          acc = wmma_f32_k4(a, bv, acc);
        }
        #pragma unroll
        for (int r = 0; r < 8; ++r) {
          const int row = rt * 16 + r + half * 8;
          const int col = ct * 16 + lr;
          dst[row][col] = fast_tanh(acc[r] + biasS[col]) + src[row][col];
        }
      }
    }
    __syncthreads();
  }

  // --- final projection 64 -> 3 (result lives in hB after d=3) ---
  #pragma unroll
  for (int dd = 0; dd < DIMX; ++dd) {
    float acc = final_b[dd];
    for (int k = 0; k < H1; ++k) acc += hB[tid][k] * final_w[dd * H1 + k];
    out[(b * NP + tid) * 3 + dd] = acc;
  }
}

extern "C" void kernel_launch(void* const* d_in, const int* in_sizes, int n_in,
                              void* d_out, int out_size, void* d_ws,
                              size_t ws_size, hipStream_t stream) {
  const float* x       = (const float*)d_in[0];
  const float* t       = (const float*)d_in[1];
  const float* fc1_w0  = (const float*)d_in[2];
  const float* fc1_b0  = (const float*)d_in[3];
  const float* fc1_w   = (const float*)d_in[4];
  const float* fc1_b   = (const float*)d_in[5];
  const float* fc2_w0  = (const float*)d_in[6];
  const float* fc2_b0  = (const float*)d_in[7];
  const float* fc2_w   = (const float*)d_in[8];
  const float* fc2_b   = (const float*)d_in[9];
  const float* final_w = (const float*)d_in[10];
  const float* final_b = (const float*)d_in[11];
  float* out = (float*)d_out;

  // Workspace: per-(b,i) stage means. 256*64*(4+3*32) floats = 6.55 MB.
  float* ws    = (float*)d_ws;
  float* mfeat = ws;
  float* m0    = mfeat + BATCHN * NP * 4;
  float* m1    = m0 + BATCHN * NP * H2;
  float* m2    = m1 + BATCHN * NP * H2;
  (void)in_sizes; (void)n_in; (void)out_size; (void)ws_size;

  drift_pair_chain<<<dim3(NP, BATCHN), 64, 0, stream>>>(
      x, t, fc2_w0, fc2_b0, fc2_w, fc2_b, mfeat, m0, m1, m2);
  drift_particle_chain<<<dim3(BATCHN), 64, 0, stream>>>(
      x, t, fc1_w0, fc1_b0, fc1_w, fc1_b, final_w, final_b,
      mfeat, m0, m1, m2, out);
}